// SlotAttentionVideo_31920196944329
// MI455X (gfx1250) — compile-verified
//
#include <hip/hip_runtime.h>

// ---------------- problem constants ----------------
#define BB   16
#define TT   12
#define NN   1024
#define DD   256
#define SS   128
#define KK   8
#define HH   256      // slot-MLP hidden
#define FF   512      // predictor FFN hidden
#define NHEAD 4
#define DH   32
#define EPSA 1e-8f

typedef float v2f __attribute__((ext_vector_type(2)));
typedef float v8f __attribute__((ext_vector_type(8)));

__device__ __forceinline__ v8f v8zero() {
  v8f z;
#pragma unroll
  for (int i = 0; i < 8; ++i) z[i] = 0.f;
  return z;
}

// CDNA5 fp32 WMMA: D(16x16) = A(16x4) * B(4x16) + C
__device__ __forceinline__ v8f wmma4(v2f a, v2f b, v8f c) {
  return __builtin_amdgcn_wmma_f32_16x16x4_f32(false, a, false, b, (short)0, c,
                                               false, false);
}

// per-row LN stats for 8 rows of length 128 stored contiguously in LDS
__device__ __forceinline__ void row_stats_8x128(const float* rows,
                                                float (*mrs)[2], int tid) {
  if (tid < 8) {
    float s = 0.f, q = 0.f;
    for (int d = 0; d < 128; ++d) {
      float v = rows[tid * 128 + d];
      s += v; q += v * v;
    }
    float m = s * (1.f / 128.f);
    float var = q * (1.f / 128.f) - m * m;
    mrs[tid][0] = m;
    mrs[tid][1] = rsqrtf(var + 1e-5f);
  }
}

// =====================================================================
// K0: slots0 = mu + exp(log_sigma) * noise       (B,K,S)
// =====================================================================
__global__ void k_slots_init(const float* __restrict__ noise,
                             const float* __restrict__ mu,
                             const float* __restrict__ logsig,
                             float* __restrict__ slots) {
  int i = blockIdx.x * 256 + threadIdx.x;
  if (i < BB * KK * SS) {
    int s = i & (SS - 1);
    slots[i] = mu[s] + __expf(logsig[s]) * noise[i];
  }
}

// =====================================================================
// K1: x = LN(inputs); k = (x@Wk)*S^-0.5 ; v = x@Wv   (fused, WMMA f32)
// grid = (B*T*N)/32 blocks, 256 threads (8 waves). 32 tokens per block.
// =====================================================================
#define XPAD 260
__global__ __launch_bounds__(256) void k_ln_kv(
    const float* __restrict__ x, const float* __restrict__ lnw,
    const float* __restrict__ lnb, const float* __restrict__ Wk,
    const float* __restrict__ Wv, float* __restrict__ kbuf,
    float* __restrict__ vbuf) {
  __shared__ float xn[32 * XPAD];
  __shared__ float redS[32][8];
  __shared__ float redQ[32][8];
  __shared__ float mrs[32][2];
  const int tid = threadIdx.x;
  const size_t row0 = (size_t)blockIdx.x * 32;

  // stage raw rows
  for (int i = tid; i < 32 * DD; i += 256) {
    int r = i >> 8, d = i & 255;
    xn[r * XPAD + d] = x[(row0 + r) * DD + d];
  }
  __syncthreads();
  // stats: 8 partials per row
  {
    int r = tid >> 3, p = tid & 7;
    float s = 0.f, q = 0.f;
    for (int d = p; d < DD; d += 8) {
      float v = xn[r * XPAD + d];
      s += v; q += v * v;
    }
    redS[r][p] = s; redQ[r][p] = q;
  }
  __syncthreads();
  if (tid < 32) {
    float s = 0.f, q = 0.f;
    for (int p = 0; p < 8; ++p) { s += redS[tid][p]; q += redQ[tid][p]; }
    float m = s * (1.f / 256.f);
    float var = q * (1.f / 256.f) - m * m;
    mrs[tid][0] = m;
    mrs[tid][1] = rsqrtf(var + 1e-5f);
  }
  __syncthreads();
  for (int i = tid; i < 32 * DD; i += 256) {
    int r = i >> 8, d = i & 255;
    xn[r * XPAD + d] =
        (xn[r * XPAD + d] - mrs[r][0]) * mrs[r][1] * lnw[d] + lnb[d];
  }
  __syncthreads();

  // WMMA GEMM: 2 M-tiles x 8 N-tiles x 2 matrices = 32 tiles over 8 waves
  const int wave = tid >> 5, lane = tid & 31;
  const int lrow = lane & 15, kh = lane >> 4;
  for (int tile = wave; tile < 32; tile += 8) {
    int mat = tile >> 4, rest = tile & 15;
    int mt = rest >> 3, nt = rest & 7;
    const float* W = mat ? Wv : Wk;
    float* dst = mat ? vbuf : kbuf;
    int arow = (mt * 16 + lrow) * XPAD;
    int bcol = nt * 16 + lrow;
    v8f c = v8zero();
    for (int k0 = 0; k0 < DD; k0 += 4) {
      int kk = k0 + 2 * kh;
      v2f a; a.x = xn[arow + kk]; a.y = xn[arow + kk + 1];
      v2f b; b.x = W[kk * SS + bcol]; b.y = W[(kk + 1) * SS + bcol];
      c = wmma4(a, b, c);
    }
    if (mat == 0) {
#pragma unroll
      for (int r = 0; r < 8; ++r) c[r] *= 0.08838834764831845f;  // S^-0.5
    }
#pragma unroll
    for (int r = 0; r < 8; ++r) {
      size_t grow = row0 + (size_t)(mt * 16 + r + 8 * kh);
      dst[grow * SS + bcol] = c[r];
    }
  }
}

// =====================================================================
// K2: q = LN(slots, ln_slots) @ Wq                (B,K,S)
// grid = B blocks, 128 threads.
// =====================================================================
__global__ __launch_bounds__(128) void k_q(
    const float* __restrict__ slots, const float* __restrict__ lnw,
    const float* __restrict__ lnb, const float* __restrict__ Wq,
    float* __restrict__ qbuf) {
  __shared__ float lns[KK * SS];
  __shared__ float mrs[8][2];
  int b = blockIdx.x, tid = threadIdx.x;
  const float* sb = slots + b * KK * SS;
  for (int i = tid; i < KK * SS; i += 128) lns[i] = sb[i];
  __syncthreads();
  row_stats_8x128(lns, mrs, tid);
  __syncthreads();
  for (int k = 0; k < KK; ++k)
    lns[k * SS + tid] =
        (lns[k * SS + tid] - mrs[k][0]) * mrs[k][1] * lnw[tid] + lnb[tid];
  __syncthreads();
  float acc[KK];
#pragma unroll
  for (int k = 0; k < KK; ++k) acc[k] = 0.f;
  for (int d = 0; d < SS; ++d) {
    float wv = Wq[d * SS + tid];
#pragma unroll
    for (int k = 0; k < KK; ++k) acc[k] += lns[k * SS + d] * wv;
  }
#pragma unroll
  for (int k = 0; k < KK; ++k) qbuf[b * KK * SS + k * SS + tid] = acc[k];
}

// =====================================================================
// K3: logits = k_t @ q^T, softmax over slots, attn_vis out, attn+EPS buf
// grid = B*8 blocks (128 tokens each), 256 threads (8 waves, WMMA)
// =====================================================================
#define QPAD 132
#define LPAD 17
__global__ __launch_bounds__(256) void k_logits(
    const float* __restrict__ kbuf, const float* __restrict__ qbuf,
    float* __restrict__ attnbuf, float* __restrict__ attn_out, int t,
    int write_out) {
  __shared__ float qs[16 * QPAD];
  __shared__ float lg[128 * LPAD];
  int blk = blockIdx.x;
  int b = blk >> 3, chunk = blk & 7;
  int tid = threadIdx.x;
  for (int i = tid; i < 16 * QPAD; i += 256) qs[i] = 0.f;
  __syncthreads();
  for (int i = tid; i < KK * SS; i += 256) {
    int sl = i >> 7, d = i & 127;
    qs[sl * QPAD + d] = qbuf[b * KK * SS + i];
  }
  __syncthreads();

  const int wave = tid >> 5, lane = tid & 31;
  const int lrow = lane & 15, kh = lane >> 4;
  const float* kt = kbuf + (size_t)(b * TT + t) * NN * SS;
  {
    v8f c = v8zero();
    int base = (chunk * 128 + wave * 16 + lrow) * SS;
    for (int d0 = 0; d0 < SS; d0 += 4) {
      int dd = d0 + 2 * kh;
      v2f a; a.x = kt[base + dd]; a.y = kt[base + dd + 1];
      v2f bb; bb.x = qs[lrow * QPAD + dd]; bb.y = qs[lrow * QPAD + dd + 1];
      c = wmma4(a, bb, c);
    }
#pragma unroll
    for (int r = 0; r < 8; ++r)
      lg[(wave * 16 + r + 8 * kh) * LPAD + lrow] = c[r];
  }
  __syncthreads();

  if (tid < 128) {
    int n = chunk * 128 + tid;
    float v[KK];
    float m = -1e30f;
#pragma unroll
    for (int k = 0; k < KK; ++k) { v[k] = lg[tid * LPAD + k]; m = fmaxf(m, v[k]); }
    float s = 0.f;
#pragma unroll
    for (int k = 0; k < KK; ++k) { v[k] = __expf(v[k] - m); s += v[k]; }
    float inv = 1.f / s;
    float* ab = attnbuf + (size_t)(b * NN + n) * 16;
#pragma unroll
    for (int k = 0; k < KK; ++k) {
      float av = v[k] * inv;
      if (write_out)
        attn_out[((size_t)(b * TT + t) * NN + n) * KK + k] = av;
      ab[k] = av + EPSA;
    }
#pragma unroll
    for (int k = KK; k < 16; ++k) ab[k] = 0.f;
  }
}

// =====================================================================
// K4: colsum (deterministic) + updates = (attn/colsum)^T @ v_t   (WMMA)
// grid = B blocks, 256 threads (8 waves, one 16-col s-tile each)
// =====================================================================
__global__ __launch_bounds__(256) void k_updates(
    const float* __restrict__ attnbuf, const float* __restrict__ vbuf,
    float* __restrict__ updates, int t) {
  __shared__ float red[256];
  __shared__ float cs[KK];
  int b = blockIdx.x, tid = threadIdx.x;
  const float* ab = attnbuf + (size_t)b * NN * 16;
  // deterministic column sums over N
  for (int k = 0; k < KK; ++k) {
    float p = 0.f;
    for (int n = tid; n < NN; n += 256) p += ab[n * 16 + k];
    red[tid] = p;
    __syncthreads();
    for (int st = 128; st > 0; st >>= 1) {
      if (tid < st) red[tid] += red[tid + st];
      __syncthreads();
    }
    if (tid == 0) cs[k] = 1.f / red[0];
    __syncthreads();
  }

  const int wave = tid >> 5, lane = tid & 31;
  const int lrow = lane & 15, kh = lane >> 4;
  const float* vt = vbuf + (size_t)(b * TT + t) * NN * SS;
  int s = wave * 16 + lrow;
  v8f c = v8zero();
  for (int n0 = 0; n0 < NN; n0 += 4) {
    int nn = n0 + 2 * kh;
    v2f a; a.x = ab[nn * 16 + lrow]; a.y = ab[(nn + 1) * 16 + lrow];
    v2f bb; bb.x = vt[nn * SS + s]; bb.y = vt[(nn + 1) * SS + s];
    c = wmma4(a, bb, c);
  }
  if (kh == 0) {
#pragma unroll
    for (int r = 0; r < 8; ++r)
      updates[b * KK * SS + r * SS + s] = c[r] * cs[r];
  }
}

// =====================================================================
// K5: gi = updates@Wih^T + bih ; gh = slots@Whh^T + bhh   (WMMA)
// M = B*K = 128 rows, N = 384, K = 128.  grid = 48 blocks, 256 threads.
// =====================================================================
__global__ __launch_bounds__(256) void k_gru_gemm(
    const float* __restrict__ updates, const float* __restrict__ slots,
    const float* __restrict__ Wih, const float* __restrict__ Whh,
    const float* __restrict__ bih, const float* __restrict__ bhh,
    float* __restrict__ gi, float* __restrict__ gh) {
  int tid = threadIdx.x;
  const int wave = tid >> 5, lane = tid & 31;
  const int lrow = lane & 15, kh = lane >> 4;
  int tile = blockIdx.x * 8 + wave;  // 0..383
  int mat = tile / 192;
  int rest = tile % 192;
  int mt = rest / 24, nt = rest % 24;
  const float* A = mat ? slots : updates;
  const float* W = mat ? Whh : Wih;
  const float* bias = mat ? bhh : bih;
  float* dst = mat ? gh : gi;
  int arow = (mt * 16 + lrow) * SS;
  int j = nt * 16 + lrow;
  v8f c = v8zero();
  for (int d0 = 0; d0 < SS; d0 += 4) {
    int dd = d0 + 2 * kh;
    v2f a; a.x = A[arow + dd]; a.y = A[arow + dd + 1];
    v2f bb; bb.x = W[j * SS + dd]; bb.y = W[j * SS + dd + 1];
    c = wmma4(a, bb, c);
  }
  float bj = bias[j];
#pragma unroll
  for (int r = 0; r < 8; ++r)
    dst[(mt * 16 + r + 8 * kh) * 384 + j] = c[r] + bj;
}

// =====================================================================
// K6: GRU gates (elementwise) + optional write of slots_collect
// =====================================================================
__global__ __launch_bounds__(256) void k_gru_gate(
    const float* __restrict__ gi, const float* __restrict__ gh,
    float* __restrict__ slots, float* __restrict__ slots_out, int t,
    int last) {
  int idx = blockIdx.x * 256 + threadIdx.x;
  if (idx >= BB * KK * SS) return;
  int row = idx >> 7, s = idx & 127;
  float ir = gi[row * 384 + s], iz = gi[row * 384 + 128 + s],
        inn = gi[row * 384 + 256 + s];
  float hr = gh[row * 384 + s], hz = gh[row * 384 + 128 + s],
        hn = gh[row * 384 + 256 + s];
  float r = 1.f / (1.f + __expf(-(ir + hr)));
  float z = 1.f / (1.f + __expf(-(iz + hz)));
  float n = tanhf(inn + r * hn);
  float h = slots[idx];
  float nh = (1.f - z) * n + z * h;
  slots[idx] = nh;
  if (last) {
    int b = row >> 3, k = row & 7;
    slots_out[((size_t)(b * TT + t) * KK + k) * SS + s] = nh;
  }
}

// =====================================================================
// K7: slot MLP residual: slots += relu(LN(slots)@W1+b1)@W2+b2
// grid = B blocks, 128 threads.
// =====================================================================
__global__ __launch_bounds__(128) void k_mlp(
    float* __restrict__ slots, const float* __restrict__ lnw,
    const float* __restrict__ lnb, const float* __restrict__ W1,
    const float* __restrict__ b1, const float* __restrict__ W2,
    const float* __restrict__ b2) {
  __shared__ float lns[KK * SS];
  __shared__ float hid[KK * HH];
  __shared__ float mrs[8][2];
  int b = blockIdx.x, tid = threadIdx.x;
  float* sb = slots + b * KK * SS;
  for (int i = tid; i < KK * SS; i += 128) lns[i] = sb[i];
  __syncthreads();
  row_stats_8x128(lns, mrs, tid);
  __syncthreads();
  for (int k = 0; k < KK; ++k)
    lns[k * SS + tid] =
        (lns[k * SS + tid] - mrs[k][0]) * mrs[k][1] * lnw[tid] + lnb[tid];
  __syncthreads();
  for (int half = 0; half < 2; ++half) {
    int col = tid + 128 * half;
    for (int k = 0; k < KK; ++k) {
      float acc = b1[col];
      for (int d = 0; d < SS; ++d) acc += lns[k * SS + d] * W1[d * HH + col];
      hid[k * HH + col] = fmaxf(acc, 0.f);
    }
  }
  __syncthreads();
  for (int k = 0; k < KK; ++k) {
    float acc = b2[tid];
    for (int h2 = 0; h2 < HH; ++h2) acc += hid[k * HH + h2] * W2[h2 * SS + tid];
    sb[k * SS + tid] += acc;
  }
}

// =====================================================================
// Predictor stage 1: y = LN(slots,p_ln1); pq=(y@Wq)*dh^-.5; pk=y@Wk; pv=y@Wv
// =====================================================================
__global__ __launch_bounds__(128) void k_pqkv(
    const float* __restrict__ slots, const float* __restrict__ lnw,
    const float* __restrict__ lnb, const float* __restrict__ Wq,
    const float* __restrict__ Wk, const float* __restrict__ Wv,
    float* __restrict__ pq, float* __restrict__ pk, float* __restrict__ pv) {
  __shared__ float lns[KK * SS];
  __shared__ float mrs[8][2];
  int b = blockIdx.x, tid = threadIdx.x;
  const float* sb = slots + b * KK * SS;
  for (int i = tid; i < KK * SS; i += 128) lns[i] = sb[i];
  __syncthreads();
  row_stats_8x128(lns, mrs, tid);
  __syncthreads();
  for (int k = 0; k < KK; ++k)
    lns[k * SS + tid] =
        (lns[k * SS + tid] - mrs[k][0]) * mrs[k][1] * lnw[tid] + lnb[tid];
  __syncthreads();
  float aq[KK], ak[KK], av[KK];
#pragma unroll
  for (int k = 0; k < KK; ++k) { aq[k] = 0.f; ak[k] = 0.f; av[k] = 0.f; }
  for (int d = 0; d < SS; ++d) {
    float wq = Wq[d * SS + tid], wk = Wk[d * SS + tid], wv = Wv[d * SS + tid];
#pragma unroll
    for (int k = 0; k < KK; ++k) {
      float l = lns[k * SS + d];
      aq[k] += l * wq; ak[k] += l * wk; av[k] += l * wv;
    }
  }
#pragma unroll
  for (int k = 0; k < KK; ++k) {
    pq[b * KK * SS + k * SS + tid] = aq[k] * 0.17677669529663687f;  // dh^-0.5
    pk[b * KK * SS + k * SS + tid] = ak[k];
    pv[b * KK * SS + k * SS + tid] = av[k];
  }
}

// =====================================================================
// Predictor stage 2: 4-head attention over K=8 slots -> po (pre-Wo)
// =====================================================================
__global__ __launch_bounds__(128) void k_pattn(const float* __restrict__ pq,
                                               const float* __restrict__ pk,
                                               const float* __restrict__ pv,
                                               float* __restrict__ po) {
  __shared__ float lq[KK * SS], lk[KK * SS], lv[KK * SS];
  __shared__ float a[NHEAD * KK * KK];
  int b = blockIdx.x, tid = threadIdx.x;
  for (int i = tid; i < KK * SS; i += 128) {
    lq[i] = pq[b * KK * SS + i];
    lk[i] = pk[b * KK * SS + i];
    lv[i] = pv[b * KK * SS + i];
  }
  __syncthreads();
  for (int e = tid; e < NHEAD * KK * KK; e += 128) {
    int h = e >> 6, qk = (e >> 3) & 7, kk2 = e & 7;
    float acc = 0.f;
    for (int j = 0; j < DH; ++j)
      acc += lq[qk * SS + h * DH + j] * lk[kk2 * SS + h * DH + j];
    a[e] = acc;
  }
  __syncthreads();
  if (tid < NHEAD * KK) {
    int h = tid >> 3, qk = tid & 7;
    float* row = &a[h * 64 + qk * 8];
    float m = -1e30f;
    for (int k2 = 0; k2 < KK; ++k2) m = fmaxf(m, row[k2]);
    float s = 0.f;
    for (int k2 = 0; k2 < KK; ++k2) { row[k2] = __expf(row[k2] - m); s += row[k2]; }
    float inv = 1.f / s;
    for (int k2 = 0; k2 < KK; ++k2) row[k2] *= inv;
  }
  __syncthreads();
  int h = tid >> 5, j = tid & 31;
  for (int ko = 0; ko < KK; ++ko) {
    float acc = 0.f;
    for (int kk2 = 0; kk2 < KK; ++kk2)
      acc += a[h * 64 + ko * 8 + kk2] * lv[kk2 * SS + h * DH + j];
    po[b * KK * SS + ko * SS + h * DH + j] = acc;
  }
}

// =====================================================================
// Predictor stage 3: s1 = slots + po @ p_Wo
// =====================================================================
__global__ __launch_bounds__(128) void k_pproj(const float* __restrict__ po,
                                               const float* __restrict__ Wo,
                                               const float* __restrict__ slots,
                                               float* __restrict__ s1) {
  __shared__ float lo[KK * SS];
  int b = blockIdx.x, tid = threadIdx.x;
  for (int i = tid; i < KK * SS; i += 128) lo[i] = po[b * KK * SS + i];
  __syncthreads();
  float acc[KK];
#pragma unroll
  for (int k = 0; k < KK; ++k) acc[k] = 0.f;
  for (int d = 0; d < SS; ++d) {
    float w = Wo[d * SS + tid];
#pragma unroll
    for (int k = 0; k < KK; ++k) acc[k] += lo[k * SS + d] * w;
  }
#pragma unroll
  for (int k = 0; k < KK; ++k)
    s1[b * KK * SS + k * SS + tid] =
        slots[b * KK * SS + k * SS + tid] + acc[k];
}

// =====================================================================
// Predictor stage 4: h=LN(s1,p_ln2); s2=s1+relu(h@W1+b1)@W2+b2;
//                    slots(carry) = LN(s2, p_lnf)
// =====================================================================
__global__ __launch_bounds__(128) void k_pffn(
    const float* __restrict__ s1, const float* __restrict__ ln2w,
    const float* __restrict__ ln2b, const float* __restrict__ W1,
    const float* __restrict__ b1, const float* __restrict__ W2,
    const float* __restrict__ b2, const float* __restrict__ lnfw,
    const float* __restrict__ lnfb, float* __restrict__ slots) {
  __shared__ float lns[KK * SS];
  __shared__ float hid[KK * FF];
  __shared__ float s2l[KK * SS];
  __shared__ float mrs[8][2];
  int b = blockIdx.x, tid = threadIdx.x;
  const float* sb = s1 + b * KK * SS;
  for (int i = tid; i < KK * SS; i += 128) lns[i] = sb[i];
  __syncthreads();
  row_stats_8x128(lns, mrs, tid);
  __syncthreads();
  for (int k = 0; k < KK; ++k)
    lns[k * SS + tid] =
        (lns[k * SS + tid] - mrs[k][0]) * mrs[k][1] * ln2w[tid] + ln2b[tid];
  __syncthreads();
  for (int q4 = 0; q4 < 4; ++q4) {
    int col = tid + 128 * q4;
    for (int k = 0; k < KK; ++k) {
      float acc = b1[col];
      for (int d = 0; d < SS; ++d) acc += lns[k * SS + d] * W1[d * FF + col];
      hid[k * FF + col] = fmaxf(acc, 0.f);
    }
  }
  __syncthreads();
  for (int k = 0; k < KK; ++k) {
    float acc = b2[tid];
    for (int f = 0; f < FF; ++f) acc += hid[k * FF + f] * W2[f * SS + tid];
    s2l[k * SS + tid] = sb[k * SS + tid] + acc;
  }
  __syncthreads();
  row_stats_8x128(s2l, mrs, tid);
  __syncthreads();
  for (int k = 0; k < KK; ++k)
    slots[b * KK * SS + k * SS + tid] =
        (s2l[k * SS + tid] - mrs[k][0]) * mrs[k][1] * lnfw[tid] + lnfb[tid];
}

// =====================================================================
// host-side orchestration
// =====================================================================
extern "C" void kernel_launch(void* const* d_in, const int* in_sizes, int n_in,
                              void* d_out, int out_size, void* d_ws,
                              size_t ws_size, hipStream_t stream) {
  const float* in_x      = (const float*)d_in[0];
  const float* in_noise  = (const float*)d_in[1];
  const float* in_mu     = (const float*)d_in[2];
  const float* in_logsig = (const float*)d_in[3];
  const float* ln_in_w   = (const float*)d_in[4];
  const float* ln_in_b   = (const float*)d_in[5];
  const float* ln_sl_w   = (const float*)d_in[6];
  const float* ln_sl_b   = (const float*)d_in[7];
  const float* ln_ml_w   = (const float*)d_in[8];
  const float* ln_ml_b   = (const float*)d_in[9];
  const float* Wq        = (const float*)d_in[10];
  const float* Wk        = (const float*)d_in[11];
  const float* Wv        = (const float*)d_in[12];
  const float* gWih      = (const float*)d_in[13];
  const float* gWhh      = (const float*)d_in[14];
  const float* gbih      = (const float*)d_in[15];
  const float* gbhh      = (const float*)d_in[16];
  const float* mW1       = (const float*)d_in[17];
  const float* mb1       = (const float*)d_in[18];
  const float* mW2       = (const float*)d_in[19];
  const float* mb2       = (const float*)d_in[20];
  const float* pln1w     = (const float*)d_in[21];
  const float* pln1b     = (const float*)d_in[22];
  const float* pWq       = (const float*)d_in[23];
  const float* pWk       = (const float*)d_in[24];
  const float* pWv       = (const float*)d_in[25];
  const float* pWo       = (const float*)d_in[26];
  const float* pln2w     = (const float*)d_in[27];
  const float* pln2b     = (const float*)d_in[28];
  const float* pfW1      = (const float*)d_in[29];
  const float* pfb1      = (const float*)d_in[30];
  const float* pfW2      = (const float*)d_in[31];
  const float* pfb2      = (const float*)d_in[32];
  const float* plnfw     = (const float*)d_in[33];
  const float* plnfb     = (const float*)d_in[34];

  float* slots_out = (float*)d_out;                       // (B,T,K,S)
  float* attn_out  = (float*)d_out + (size_t)BB * TT * KK * SS;  // (B,T,N,K)

  float* ws = (float*)d_ws;
  size_t o = 0;
  float* kbuf    = ws + o; o += (size_t)BB * TT * NN * SS;
  float* vbuf    = ws + o; o += (size_t)BB * TT * NN * SS;
  float* slots   = ws + o; o += BB * KK * SS;
  float* qbuf    = ws + o; o += BB * KK * SS;
  float* attnbuf = ws + o; o += (size_t)BB * NN * 16;
  float* updates = ws + o; o += BB * KK * SS;
  float* gi      = ws + o; o += BB * KK * 384;
  float* gh      = ws + o; o += BB * KK * 384;
  float* pq      = ws + o; o += BB * KK * SS;
  float* pk      = ws + o; o += BB * KK * SS;
  float* pv      = ws + o; o += BB * KK * SS;
  float* po      = ws + o; o += BB * KK * SS;
  float* s1      = ws + o; o += BB * KK * SS;

  k_slots_init<<<(BB * KK * SS + 255) / 256, 256, 0, stream>>>(
      in_noise, in_mu, in_logsig, slots);
  k_ln_kv<<<(BB * TT * NN) / 32, 256, 0, stream>>>(in_x, ln_in_w, ln_in_b, Wk,
                                                   Wv, kbuf, vbuf);

  for (int t = 0; t < TT; ++t) {
    for (int it = 0; it < 3; ++it) {
      int last = (it == 2) ? 1 : 0;
      k_q<<<BB, 128, 0, stream>>>(slots, ln_sl_w, ln_sl_b, Wq, qbuf);
      k_logits<<<BB * 8, 256, 0, stream>>>(kbuf, qbuf, attnbuf, attn_out, t,
                                           last);
      k_updates<<<BB, 256, 0, stream>>>(attnbuf, vbuf, updates, t);
      k_gru_gemm<<<48, 256, 0, stream>>>(updates, slots, gWih, gWhh, gbih,
                                         gbhh, gi, gh);
      k_gru_gate<<<(BB * KK * SS) / 256, 256, 0, stream>>>(gi, gh, slots,
                                                           slots_out, t, last);
      if (!last)
        k_mlp<<<BB, 128, 0, stream>>>(slots, ln_ml_w, ln_ml_b, mW1, mb1, mW2,
                                      mb2);
    }
    // predictor -> new carry slots
    k_pqkv<<<BB, 128, 0, stream>>>(slots, pln1w, pln1b, pWq, pWk, pWv, pq, pk,
                                   pv);
    k_pattn<<<BB, 128, 0, stream>>>(pq, pk, pv, po);
    k_pproj<<<BB, 128, 0, stream>>>(po, pWo, slots, s1);
    k_pffn<<<BB, 128, 0, stream>>>(s1, pln2w, pln2b, pfW1, pfb1, pfW2, pfb2,
                                   plnfw, plnfb, slots);
  }
}